// RGCNRec_59725815218517
// MI455X (gfx1250) — compile-verified
//
#include <hip/hip_runtime.h>

// ---------------------------------------------------------------------------
// CDNA5 (gfx1250) 2-layer bipartite RGCN recommender.
//  - GEMMs: V_WMMA_F32_16X16X4_F32, one wave32 per 16x16 tile, K=128.
//  - Aggregation: CSR built once (int atomics + LDS scan), then pull-mode
//    register accumulation per destination node: zero float atomics,
//    b128/b64-coalesced gathers that hit MI455X's 192MB L2 (wh_u=51MB,
//    wh_i=26MB are L2-resident).
// ---------------------------------------------------------------------------

typedef __attribute__((ext_vector_type(2))) float v2f;
typedef __attribute__((ext_vector_type(8))) float v8f;

__global__ void zero_i32(int* __restrict__ p, long n) {
  long i = (long)blockIdx.x * blockDim.x + threadIdx.x;
  long stride = (long)gridDim.x * blockDim.x;
  for (; i < n; i += stride) p[i] = 0;
}

__global__ void copy_i32(const int* __restrict__ s, int* __restrict__ d, int n) {
  int i = blockIdx.x * blockDim.x + threadIdx.x;
  int stride = gridDim.x * blockDim.x;
  for (; i < n; i += stride) d[i] = s[i];
}

__global__ void count_degrees(const int* __restrict__ eu, const int* __restrict__ ei,
                              int* __restrict__ cnt_u, int* __restrict__ cnt_i, int ne) {
  int e = blockIdx.x * blockDim.x + threadIdx.x;
  if (e < ne) {
    atomicAdd(&cnt_u[eu[e]], 1);
    atomicAdd(&cnt_i[ei[e]], 1);
  }
}

// Single-workgroup exclusive scan (Hillis-Steele tiles of 1024 via LDS).
__global__ __launch_bounds__(1024)
void scan_exclusive(const int* __restrict__ cnt, int* __restrict__ off, int n) {
  __shared__ int tmp[1024];
  __shared__ int carry;
  const int tid = threadIdx.x;
  if (tid == 0) carry = 0;
  __syncthreads();
  for (int base = 0; base < n; base += 1024) {
    int i = base + tid;
    int v = (i < n) ? cnt[i] : 0;
    tmp[tid] = v;
    __syncthreads();
    for (int ofs = 1; ofs < 1024; ofs <<= 1) {
      int t = (tid >= ofs) ? tmp[tid - ofs] : 0;
      __syncthreads();
      tmp[tid] += t;
      __syncthreads();
    }
    int excl = carry + tmp[tid] - v;   // reads old carry
    if (i < n) off[i] = excl;
    int total = tmp[1023];
    __syncthreads();                   // everyone done reading carry/tmp
    if (tid == 0) carry += total;
    __syncthreads();                   // carry visible before next tile
  }
  if (tid == 0) off[n] = carry;
}

// Group source node ids by destination: list_i[] = user srcs grouped by item,
// list_u[] = item srcs grouped by user.
__global__ void fill_csr(const int* __restrict__ eu, const int* __restrict__ ei,
                         int* __restrict__ cur_u, int* __restrict__ cur_i,
                         int* __restrict__ list_u, int* __restrict__ list_i, int ne) {
  int e = blockIdx.x * blockDim.x + threadIdx.x;
  if (e >= ne) return;
  int u = eu[e], it = ei[e];
  int p = atomicAdd(&cur_i[it], 1);
  list_i[p] = u;
  int q = atomicAdd(&cur_u[u], 1);
  list_u[q] = it;
}

// C[M,N] = A[M,128] @ W[128,N] + bias[N]; one wave32 per 16x16 tile.
__global__ __launch_bounds__(128)
void gemm_wmma_f32(const float* __restrict__ A, const float* __restrict__ W,
                   const float* __restrict__ bias, float* __restrict__ C, int N) {
  const int K = 128;
  const int lane  = threadIdx.x & 31;
  const int wave  = threadIdx.x >> 5;
  const int mt    = blockIdx.x;
  const int arow  = lane & 15;
  const int khalf = (lane >> 4) << 1;          // lanes 16-31 hold K+2/K+3
  const float* Arow = A + ((long)mt * 16 + arow) * K;

  const int ntiles = N >> 4;
  for (int nt = wave; nt < ntiles; nt += 4) {
    const int col = nt * 16 + (lane & 15);
    v8f acc = {};
    for (int k0 = 0; k0 < K; k0 += 4) {
      v2f a, b;
      a.x = Arow[k0 + khalf];
      a.y = Arow[k0 + khalf + 1];
      b.x = W[(long)(k0 + khalf) * N + col];
      b.y = W[(long)(k0 + khalf + 1) * N + col];
      acc = __builtin_amdgcn_wmma_f32_16x16x4_f32(false, a, false, b,
                                                  (short)0, acc, false, false);
    }
    const float bv = bias[col];
    const int rbase = mt * 16 + ((lane >> 4) << 3);
#pragma unroll
    for (int r = 0; r < 8; ++r)
      C[(long)(rbase + r) * N + col] = acc[r] + bv;
  }
}

// Pull-mode mean aggregation, d=128: one wave per dst node; lane owns a
// float4 (16B) slice -> each edge's row is one fully-coalesced 512B b128 wave load.
__global__ __launch_bounds__(256)
void aggregate_mean_128(const int* __restrict__ offs, const int* __restrict__ srcs,
                        const float* __restrict__ src_feat, float* __restrict__ dst_feat,
                        int n_nodes, int leaky) {
  const int lane = threadIdx.x & 31;
  const int node = blockIdx.x * (blockDim.x >> 5) + (threadIdx.x >> 5);
  if (node >= n_nodes) return;
  const int s = offs[node], e = offs[node + 1];
  float a0 = 0.f, a1 = 0.f, a2 = 0.f, a3 = 0.f;
  for (int p = s; p < e; ++p) {
    const float4 v = ((const float4*)(src_feat + (long)srcs[p] * 128))[lane];
    a0 += v.x; a1 += v.y; a2 += v.z; a3 += v.w;
  }
  const float inv = 1.0f / fmaxf((float)(e - s), 1.0f);
  a0 *= inv; a1 *= inv; a2 *= inv; a3 *= inv;
  if (leaky) {
    a0 = a0 > 0.f ? a0 : 0.01f * a0;
    a1 = a1 > 0.f ? a1 : 0.01f * a1;
    a2 = a2 > 0.f ? a2 : 0.01f * a2;
    a3 = a3 > 0.f ? a3 : 0.01f * a3;
  }
  float4 r; r.x = a0; r.y = a1; r.z = a2; r.w = a3;
  ((float4*)(dst_feat + (long)node * 128))[lane] = r;
}

// Same, d=64: float2 per lane (256B b64 wave load per edge row).
__global__ __launch_bounds__(256)
void aggregate_mean_64(const int* __restrict__ offs, const int* __restrict__ srcs,
                       const float* __restrict__ src_feat, float* __restrict__ dst_feat,
                       int n_nodes, int leaky) {
  const int lane = threadIdx.x & 31;
  const int node = blockIdx.x * (blockDim.x >> 5) + (threadIdx.x >> 5);
  if (node >= n_nodes) return;
  const int s = offs[node], e = offs[node + 1];
  float a0 = 0.f, a1 = 0.f;
  for (int p = s; p < e; ++p) {
    const float2 v = ((const float2*)(src_feat + (long)srcs[p] * 64))[lane];
    a0 += v.x; a1 += v.y;
  }
  const float inv = 1.0f / fmaxf((float)(e - s), 1.0f);
  a0 *= inv; a1 *= inv;
  if (leaky) {
    a0 = a0 > 0.f ? a0 : 0.01f * a0;
    a1 = a1 > 0.f ? a1 : 0.01f * a1;
  }
  float2 r; r.x = a0; r.y = a1;
  ((float2*)(dst_feat + (long)node * 64))[lane] = r;
}

// One wave32 per (b,l); 64-dim vectors as float2/lane, dots via shfl_xor tree.
__global__ __launch_bounds__(256)
void score_kernel(const float* __restrict__ hu, const float* __restrict__ hi,
                  const int* __restrict__ uid, const int* __restrict__ pos,
                  const int* __restrict__ neg, float* __restrict__ out) {
  const int OUTD = 64, L = 5, Kn = 4, B = 2048;
  const int lane = threadIdx.x & 31;
  const int wid = blockIdx.x * (blockDim.x >> 5) + (threadIdx.x >> 5); // b*L + l
  if (wid >= B * L) return;
  const int b = wid / L;

  const float2 u = ((const float2*)(hu + (long)uid[b] * OUTD))[lane];
  const float2 p = ((const float2*)(hi + (long)pos[wid] * OUTD))[lane];

  float* pos_logits   = out;            // [B*L]
  float* neg_logits   = out + 10240;    // [B*L*K]
  float* ue_neg       = out + 51200;    // [B*L*K*64]
  float* pos_item_emb = out + 2672640;  // [B*L*64]
  float* neg_item_emb = out + 3328000;  // [B*L*K*64]

  ((float2*)(pos_item_emb + (long)wid * OUTD))[lane] = p;

  float dot = u.x * p.x + u.y * p.y;
#pragma unroll
  for (int off = 16; off > 0; off >>= 1) dot += __shfl_xor(dot, off, 32);
  if (lane == 0) pos_logits[wid] = dot;

  for (int k = 0; k < Kn; ++k) {
    const int idx = wid * Kn + k;
    const float2 nv = ((const float2*)(hi + (long)neg[idx] * OUTD))[lane];
    ((float2*)(neg_item_emb + (long)idx * OUTD))[lane] = nv;
    ((float2*)(ue_neg + (long)idx * OUTD))[lane] = u;
    float d2 = u.x * nv.x + u.y * nv.y;
#pragma unroll
    for (int off = 16; off > 0; off >>= 1) d2 += __shfl_xor(d2, off, 32);
    if (lane == 0) neg_logits[idx] = d2;
  }
}

extern "C" void kernel_launch(void* const* d_in, const int* in_sizes, int n_in,
                              void* d_out, int out_size, void* d_ws, size_t ws_size,
                              hipStream_t stream) {
  (void)in_sizes; (void)n_in; (void)out_size; (void)ws_size;
  const float* emb_user = (const float*)d_in[0];
  const float* emb_item = (const float*)d_in[1];
  const float* W1_ui = (const float*)d_in[2];
  const float* b1_ui = (const float*)d_in[3];
  const float* W1_iu = (const float*)d_in[4];
  const float* b1_iu = (const float*)d_in[5];
  const float* W2_ui = (const float*)d_in[6];
  const float* b2_ui = (const float*)d_in[7];
  const float* W2_iu = (const float*)d_in[8];
  const float* b2_iu = (const float*)d_in[9];
  const int* uid     = (const int*)d_in[10];
  const int* pos     = (const int*)d_in[11];
  const int* neg     = (const int*)d_in[12];
  const int* edges_u = (const int*)d_in[13];
  const int* edges_i = (const int*)d_in[14];
  float* out = (float*)d_out;

  const long NU = 100000, NI = 50000, NE = 1000000;
  const int HID = 128, OUTD = 64;

  // workspace layout: 38.4M floats (153.6MB) + ~2.3M ints (9.2MB)
  float* ws   = (float*)d_ws;
  float* bufA = ws;               // NU*128 : wh_u  / wh2_u
  float* bufB = bufA + NU * HID;  // NI*128 : wh_i  / wh2_i
  float* bufC = bufB + NI * HID;  // NU*128 : hu -> hu2
  float* bufD = bufC + NU * HID;  // NI*128 : hi -> hi2
  int* cur_u  = (int*)(bufD + NI * HID);  // NU   (count, then cursor)
  int* cur_i  = cur_u + NU;               // NI
  int* off_u  = cur_i + NI;               // NU+1
  int* off_i  = off_u + NU + 1;           // NI+1
  int* list_u = off_i + NI + 1;           // NE : item srcs grouped by user dst
  int* list_i = list_u + NE;              // NE : user srcs grouped by item dst

  // ---- build CSR once (shared by both layers & directions) ----
  zero_i32<<<512, 256, 0, stream>>>(cur_u, NU + NI);
  count_degrees<<<(int)((NE + 255) / 256), 256, 0, stream>>>(edges_u, edges_i,
                                                             cur_u, cur_i, (int)NE);
  scan_exclusive<<<1, 1024, 0, stream>>>(cur_u, off_u, (int)NU);
  scan_exclusive<<<1, 1024, 0, stream>>>(cur_i, off_i, (int)NI);
  copy_i32<<<512, 256, 0, stream>>>(off_u, cur_u, (int)NU);
  copy_i32<<<512, 256, 0, stream>>>(off_i, cur_i, (int)NI);
  fill_csr<<<(int)((NE + 255) / 256), 256, 0, stream>>>(edges_u, edges_i, cur_u, cur_i,
                                                        list_u, list_i, (int)NE);

  // ---- layer 1 ----
  gemm_wmma_f32<<<(int)(NU / 16), 128, 0, stream>>>(emb_user, W1_ui, b1_ui, bufA, HID);
  gemm_wmma_f32<<<(int)(NI / 16), 128, 0, stream>>>(emb_item, W1_iu, b1_iu, bufB, HID);
  aggregate_mean_128<<<(int)((NI + 7) / 8), 256, 0, stream>>>(off_i, list_i, bufA, bufD,
                                                              (int)NI, 1); // hi
  aggregate_mean_128<<<(int)((NU + 7) / 8), 256, 0, stream>>>(off_u, list_u, bufB, bufC,
                                                              (int)NU, 1); // hu

  // ---- layer 2 ----
  gemm_wmma_f32<<<(int)(NU / 16), 128, 0, stream>>>(bufC, W2_ui, b2_ui, bufA, OUTD);
  gemm_wmma_f32<<<(int)(NI / 16), 128, 0, stream>>>(bufD, W2_iu, b2_iu, bufB, OUTD);
  aggregate_mean_64<<<(int)((NI + 7) / 8), 256, 0, stream>>>(off_i, list_i, bufA, bufD,
                                                             (int)NI, 0); // hi2
  aggregate_mean_64<<<(int)((NU + 7) / 8), 256, 0, stream>>>(off_u, list_u, bufB, bufC,
                                                             (int)NU, 0); // hu2

  // ---- scoring / output gather ----
  score_kernel<<<(2048 * 5 + 7) / 8, 256, 0, stream>>>(bufC, bufD, uid, pos, neg, out);
}